// CubECC2d_84997402788556
// MI455X (gfx1250) — compile-verified
//
#include <hip/hip_runtime.h>

#define BATCH 128
#define CHAN 3
#define HH 256
#define WW 256
#define STEPS 64
#define NIMG (BATCH * CHAN)          /* 384 images */
#define TILE_H 32
#define CHUNKS (HH / TILE_H)         /* 8 row-chunks per image */
#define TOT_ROWS (NIMG * HH)         /* 98304 rows of 256 floats */

typedef __attribute__((ext_vector_type(4))) unsigned int v4u;
typedef __attribute__((ext_vector_type(8))) int v8i;
typedef __attribute__((ext_vector_type(4))) int v4i;
typedef __attribute__((ext_vector_type(2))) float v2f;
typedef __attribute__((ext_vector_type(8))) float v8f;

__global__ void __launch_bounds__(256) CubECC_zero_ws(int* ws, int n) {
    int i = blockIdx.x * blockDim.x + threadIdx.x;
    if (i < n) ws[i] = 0;
}

__global__ void __launch_bounds__(256) CubECC_hist(const float* __restrict__ x,
                                                   int* __restrict__ ws) {
    __shared__ float tile[(TILE_H + 1) * WW];   // 33 rows x 256 cols = 33 KB
    __shared__ int hist[8][STEPS];              // per-wave histograms, 2 KB

    const int tid = threadIdx.x;
    const int img = blockIdx.x;                 // 0..383
    const int chunk = blockIdx.y;               // 0..7
    const long long r0 = (long long)img * HH + (long long)chunk * TILE_H;

    // ---- Wave 0: issue one TDM DMA of the (32+halo)-row tile into LDS ----
    if ((tid >> 5) == 0) {
        unsigned long long ga = (unsigned long long)(const void*)(x + r0 * WW);
        unsigned int ldsa = (unsigned int)(unsigned long long)(void*)&tile[0];
        int rows = TILE_H + 1;
        if (r0 + rows > (long long)TOT_ROWS) rows = (int)((long long)TOT_ROWS - r0); // last block: 32

        v4u g0;
        g0.x = 1u;                                                   // count=1 valid, user mode
        g0.y = ldsa;                                                 // lds_addr (bytes)
        g0.z = (unsigned int)(ga & 0xFFFFFFFFull);                   // global_addr[31:0]
        g0.w = (unsigned int)((ga >> 32) & 0x1FFFFFFull) | (2u << 30); // addr[56:32] | type=2

        v8i g1;
        g1[0] = (int)(2u << 16);                                     // data_size=4B, no multicast/pad
        g1[1] = (int)((unsigned)WW << 16);                           // tensor_dim0[15:0]=256 (hi bits 0)
        g1[2] = (int)(((unsigned)TOT_ROWS & 0xFFFFu) << 16);         // dim0[31:16]=0 | dim1[15:0]
        g1[3] = (int)(((unsigned)TOT_ROWS >> 16) & 0xFFFFu) | (WW << 16); // dim1[31:16] | tile_dim0=256
        g1[4] = rows;                                                // tile_dim1 | tile_dim2=0
        g1[5] = WW;                                                  // tensor_dim0_stride[31:0]=256
        g1[6] = 0;                                                   // stride hi | dim1_stride lo (2D: unused)
        g1[7] = 0;

        v4i g2 = {0, 0, 0, 0};
        v4i g3 = {0, 0, 0, 0};
        v8i g4 = {0, 0, 0, 0, 0, 0, 0, 0};
        __builtin_amdgcn_tensor_load_to_lds(g0, g1, g2, g3, g4, 0);
    }

    // Zero per-wave histograms while the DMA is in flight (512 ints / 256 threads).
    ((int*)hist)[tid] = 0;
    ((int*)hist)[tid + 256] = 0;

    if ((tid >> 5) == 0) __builtin_amdgcn_s_wait_tensorcnt(0);
    __syncthreads();

    int* h = hist[tid >> 5];
    const int j = tid;                                   // one column per lane
    const float INV_RES = (float)(63.0 / (0.98 - 0.02)); // 1/RES

    for (int i = 0; i < TILE_H; ++i) {
        const float v00 = tile[i * WW + j];
        const float v01 = tile[i * WW + j + 1];          // j=255 reads row i+1 col 0: in-tile, guarded below
        const float v10 = tile[(i + 1) * WW + j];        // halo row; guarded below
        const float v11 = tile[(i + 1) * WW + j + 1];
        const bool right = (j < WW - 1);
        const bool down = (chunk * TILE_H + i) < (HH - 1);

        {   // vertex, sign +1
            float f = v00;
            if (f <= 0.98f) {
                int t = (int)ceilf((f - 0.02f) * INV_RES);
                atomicAdd(&h[min(max(t, 0), STEPS - 1)], 1);
            }
        }
        if (right) {   // horizontal edge, sign -1
            float f = fmaxf(v00, v01);
            if (f <= 0.98f) {
                int t = (int)ceilf((f - 0.02f) * INV_RES);
                atomicAdd(&h[min(max(t, 0), STEPS - 1)], -1);
            }
        }
        if (down) {    // vertical edge, sign -1
            float f = fmaxf(v00, v10);
            if (f <= 0.98f) {
                int t = (int)ceilf((f - 0.02f) * INV_RES);
                atomicAdd(&h[min(max(t, 0), STEPS - 1)], -1);
            }
        }
        if (right && down) {   // square, sign +1
            float f = fmaxf(fmaxf(v00, v01), fmaxf(v10, v11));
            if (f <= 0.98f) {
                int t = (int)ceilf((f - 0.02f) * INV_RES);
                atomicAdd(&h[min(max(t, 0), STEPS - 1)], 1);
            }
        }
    }

    __syncthreads();
    if (tid < STEPS) {
        int s = 0;
        #pragma unroll
        for (int w = 0; w < 8; ++w) s += hist[w][tid];
        atomicAdd(&ws[img * STEPS + tid], s);
    }
}

// Cumulative sum as triangular GEMM: OUT[384x64] = HIST[384x64] @ L[64x64],
// L[k][n] = (k <= n). One wave per block computes a 16x64 tile with
// V_WMMA_F32_16X16X4_F32 (16 K-steps x 4 N-tiles). Counts < 2^24 -> exact in f32.
__global__ void __launch_bounds__(32) CubECC_scan_wmma(const int* __restrict__ ws,
                                                       float* __restrict__ out) {
    const int lane = threadIdx.x;            // 0..31, full wave, no divergence
    const int mbase = blockIdx.x * 16;       // image rows of this tile
    const int mrow = lane & 15;              // A row / B col within lane half
    const int hi = lane >> 4;                // 0: lanes 0-15, 1: lanes 16-31
    const int kb = hi * 2;                   // A/B K-pair base per lane half

    v8f c0 = {}, c1 = {}, c2 = {}, c3 = {};

    for (int k0 = 0; k0 < STEPS; k0 += 4) {
        // A fragment: 16x4 f32. VGPR0 = K=kb, VGPR1 = K=kb+1 (kb = 0 or 2 per lane half).
        const int* hrow = ws + (mbase + mrow) * STEPS + k0 + kb;
        v2f a;
        a.x = (float)hrow[0];
        a.y = (float)hrow[1];

        // B fragment: 4x16 ones-triangular, generated per N-tile.
        const int ka = k0 + kb;
        #define ECC_BTILE(N0) ({ v2f b_;                              \
            b_.x = (ka     <= (N0) + mrow) ? 1.0f : 0.0f;             \
            b_.y = (ka + 1 <= (N0) + mrow) ? 1.0f : 0.0f; b_; })

        c0 = __builtin_amdgcn_wmma_f32_16x16x4_f32(false, a, false, ECC_BTILE(0),  (short)0, c0, false, false);
        c1 = __builtin_amdgcn_wmma_f32_16x16x4_f32(false, a, false, ECC_BTILE(16), (short)0, c1, false, false);
        c2 = __builtin_amdgcn_wmma_f32_16x16x4_f32(false, a, false, ECC_BTILE(32), (short)0, c2, false, false);
        c3 = __builtin_amdgcn_wmma_f32_16x16x4_f32(false, a, false, ECC_BTILE(48), (short)0, c3, false, false);
        #undef ECC_BTILE
    }

    // D layout: VGPR v -> row v (lanes 0-15) or v+8 (lanes 16-31), col = lane&15.
    #pragma unroll
    for (int v = 0; v < 8; ++v) {
        const int m = mbase + v + hi * 8;
        const int n = lane & 15;
        out[m * STEPS + n]      = c0[v];
        out[m * STEPS + n + 16] = c1[v];
        out[m * STEPS + n + 32] = c2[v];
        out[m * STEPS + n + 48] = c3[v];
    }
}

extern "C" void kernel_launch(void* const* d_in, const int* in_sizes, int n_in,
                              void* d_out, int out_size, void* d_ws, size_t ws_size,
                              hipStream_t stream) {
    (void)in_sizes; (void)n_in; (void)out_size; (void)ws_size;
    const float* x = (const float*)d_in[0];
    float* out = (float*)d_out;
    int* ws = (int*)d_ws;                      // NIMG*STEPS ints = 96 KB

    const int nhist = NIMG * STEPS;
    CubECC_zero_ws<<<(nhist + 255) / 256, 256, 0, stream>>>(ws, nhist);
    CubECC_hist<<<dim3(NIMG, CHUNKS), 256, 0, stream>>>(x, ws);
    CubECC_scan_wmma<<<NIMG / 16, 32, 0, stream>>>(ws, out);
}